// TimeIntegratedGAT_66159676228019
// MI455X (gfx1250) — compile-verified
//
#include <hip/hip_runtime.h>
#include <hip/hip_bf16.h>

// ---------------------------------------------------------------------------
// Time-integrated 2-layer GAT for MI455X (gfx1250, wave32).
// GEMMs use V_WMMA_F32_16X16X4_F32 (exact fp32). Layer-1 logits/segment-max
// are hoisted out of the time loop via positive homogeneity of leaky_relu.
// ---------------------------------------------------------------------------

typedef float v2f __attribute__((ext_vector_type(2)));
typedef float v8f __attribute__((ext_vector_type(8)));

#define NEG_SLOPE 0.2f

// ------------------------------ utility ------------------------------------
__global__ void k_zero_f32(float* __restrict__ p, long n) {
  long i = (long)blockIdx.x * blockDim.x + threadIdx.x;
  if (i < n) p[i] = 0.0f;
}

// ------------------------------ WMMA GEMM ----------------------------------
// H[N,64] = X[N,64] @ W[64,64]; one wave32 per 16-row slab.
// A frag (16x4 f32): lane l: m=l&15, kh=l>>4; a = {X[m][4k+2kh], X[m][4k+2kh+1]}
// B frag (4x16 f32): mirrored K mapping, N = l&15
// C/D  (16x16 f32): vgpr r -> row = r + 8*kh, col = l&15
__global__ __launch_bounds__(256) void k_gemm_n64(const float* __restrict__ X,
                                                  const float* __restrict__ W,
                                                  float* __restrict__ H,
                                                  int nrows) {
  int wave = (int)((blockIdx.x * blockDim.x + threadIdx.x) >> 5);
  int lane = threadIdx.x & 31;
  int row0 = wave * 16;
  if (row0 >= nrows) return;               // wave-uniform: EXEC stays all-ones
  int m  = lane & 15;
  int kh = lane >> 4;                      // 0 or 1
  const float* xrow = X + (size_t)(row0 + m) * 64;

  for (int nt = 0; nt < 4; ++nt) {         // 4 column tiles of 16
    int col0 = nt * 16;
    v8f acc = {};
#pragma unroll
    for (int kc = 0; kc < 16; ++kc) {      // K = 64 in chunks of 4
      int kb = kc * 4 + kh * 2;
      v2f a;
      a.x = xrow[kb];
      a.y = xrow[kb + 1];
      v2f b;
      b.x = W[(size_t)kb * 64 + col0 + m];
      b.y = W[(size_t)(kb + 1) * 64 + col0 + m];
      acc = __builtin_amdgcn_wmma_f32_16x16x4_f32(
          /*neg_a=*/false, a, /*neg_b=*/false, b,
          /*c_mod=*/(short)0, acc, /*reuse_a=*/false, /*reuse_b=*/false);
    }
#pragma unroll
    for (int r = 0; r < 8; ++r)
      H[(size_t)(row0 + r + 8 * kh) * 64 + col0 + m] = acc[r];
  }
}

// --------------------------- node-level kernels ----------------------------
// alpha_src[i] = H[i]·a_src ; alpha_dst[i] = H[i]·a_dst
__global__ void k_node_alpha(const float* __restrict__ H,
                             const float* __restrict__ a_s,
                             const float* __restrict__ a_d,
                             float* __restrict__ os, float* __restrict__ od,
                             int n) {
  int i = blockIdx.x * blockDim.x + threadIdx.x;
  if (i >= n) return;
  const float* h = H + (size_t)i * 64;
  float ss = 0.f, sd = 0.f;
#pragma unroll 4
  for (int c = 0; c < 64; ++c) {
    float v = h[c];
    ss += v * a_s[c];
    sd += v * a_d[c];
  }
  os[i] = ss;
  od[i] = sd;
}

// g = elu(t * num / (denom + 1e-16)), element-wise over N*64
__global__ void k_elu_g(const float* __restrict__ num,
                        const float* __restrict__ denom,
                        float* __restrict__ g, float t, long n64) {
  long idx = (long)blockIdx.x * blockDim.x + threadIdx.x;
  if (idx >= n64) return;
  long i = idx >> 6;
  float v = t * num[idx] / (denom[i] + 1e-16f);
  g[idx] = (v > 0.f) ? v : (__expf(v) - 1.f);
}

// d_out += hstep * num / (denom + 1e-16)
__global__ void k_out_accum(float* __restrict__ out,
                            const float* __restrict__ num,
                            const float* __restrict__ denom,
                            float hstep, long n64) {
  long idx = (long)blockIdx.x * blockDim.x + threadIdx.x;
  if (idx >= n64) return;
  long i = idx >> 6;
  out[idx] += hstep * num[idx] / (denom[i] + 1e-16f);
}

// ------------------------- segment-max via flip trick ----------------------
__device__ __forceinline__ unsigned flip_f32(float f) {
  unsigned u = __float_as_uint(f);
  return (u >> 31) ? ~u : (u | 0x80000000u);   // monotone float -> u32
}

// e[e] = leaky_relu(as[src]+ad[dst]); mflip[dst] = max(mflip[dst], flip(e))
__global__ void k_edge_logit_max(const int* __restrict__ src,
                                 const int* __restrict__ dst,
                                 const float* __restrict__ as,
                                 const float* __restrict__ ad,
                                 float* __restrict__ eout,
                                 unsigned* __restrict__ mflip, long ne) {
  long e = (long)blockIdx.x * blockDim.x + threadIdx.x;
  if (e >= ne) return;
  int d = dst[e];
  float v = as[src[e]] + ad[d];
  v = (v > 0.f) ? v : NEG_SLOPE * v;
  eout[e] = v;
  atomicMax(mflip + d, flip_f32(v));
}

// in-place: flipped u32 -> float ; never-written (0) -> 0.0 (matches isfinite fixup)
__global__ void k_finalize_max(unsigned* __restrict__ m, int n) {
  int i = blockIdx.x * blockDim.x + threadIdx.x;
  if (i >= n) return;
  unsigned u = m[i];
  float f = 0.0f;
  if (u != 0u) {
    unsigned bits = (u >> 31) ? (u ^ 0x80000000u) : ~u;
    f = __uint_as_float(bits);
  }
  ((float*)m)[i] = f;
}

// -------------------------- softmax + aggregation --------------------------
// 16 threads per edge; w = exp(escale*e[e] - mscale*m[dst]);
// denom[dst] += w (lane 0); num[dst][:] += w * H[src][:]  (float4 per lane)
__global__ void k_edge_aggr(const int* __restrict__ src,
                            const int* __restrict__ dst,
                            const float* __restrict__ eb,
                            const float* __restrict__ mv,
                            const float* __restrict__ H,
                            float* __restrict__ denom,
                            float* __restrict__ num,
                            float escale, float mscale, long ne) {
  long tid = (long)blockIdx.x * blockDim.x + threadIdx.x;
  long e = tid >> 4;
  int l = (int)(tid & 15);
  if (e >= ne) return;
  int s = src[e], d = dst[e];
  float w = __expf(escale * eb[e] - mscale * mv[d]);
  if (l == 0) atomicAdd(denom + d, w);
  const float4 hv = ((const float4*)(H + (size_t)s * 64))[l];
  float* np = num + (size_t)d * 64 + (size_t)l * 4;
  atomicAdd(np + 0, w * hv.x);
  atomicAdd(np + 1, w * hv.y);
  atomicAdd(np + 2, w * hv.z);
  atomicAdd(np + 3, w * hv.w);
}

// ------------------------------- launch ------------------------------------
static inline int cdiv(long a, int b) { return (int)((a + b - 1) / b); }

extern "C" void kernel_launch(void* const* d_in, const int* in_sizes, int n_in,
                              void* d_out, int out_size, void* d_ws, size_t ws_size,
                              hipStream_t stream) {
  const float* x   = (const float*)d_in[0];
  const int*  ei1  = (const int*)d_in[1];
  const int*  ei2  = (const int*)d_in[2];
  const float* W1  = (const float*)d_in[3];
  const float* a1s = (const float*)d_in[4];
  const float* a1d = (const float*)d_in[5];
  const float* W2  = (const float*)d_in[6];
  const float* a2s = (const float*)d_in[7];
  const float* a2d = (const float*)d_in[8];
  float* out = (float*)d_out;

  const int  N   = in_sizes[0] / 64;
  const long E   = in_sizes[1] / 2;
  const long N64 = (long)N * 64;
  const int* src1 = ei1,       *dst1 = ei1 + E;
  const int* src2 = ei2,       *dst2 = ei2 + E;

  // ---- workspace carve (floats) ----
  float* ws = (float*)d_ws;
  float* H1     = ws;            ws += N64;   // x@W1
  float* e1base = ws;            ws += E;     // layer-1 logits (t-independent)
  float* al1s   = ws;            ws += N;
  float* al1d   = ws;            ws += N;
  float* m1f    = ws;            ws += N;     // flip-max -> float
  float* g      = ws;            ws += N64;   // elu(layer-1 out)
  float* H2     = ws;            ws += N64;   // g@W2
  float* al2s   = ws;            ws += N;
  float* al2d   = ws;            ws += N;
  float* e2     = ws;            ws += E;     // layer-2 logits (per step)
  // contiguous per-step region (zeroed each step):
  float* denom1 = ws;            ws += N;
  float* m2f    = ws;            ws += N;
  float* denom2 = ws;            ws += N;
  float* num1   = ws;            ws += N64;
  float* num2   = ws;            ws += N64;
  const long perT = 3L * N + 2L * N64;

  const int B = 256;
  const int gNodes = cdiv(N, B);
  const int gN64   = cdiv(N64, B);
  const int gEdges = cdiv(E, B);
  const int gEdge16 = cdiv(E * 16, B);
  const int gGemm  = cdiv((long)cdiv(N, 16), B / 32);  // waves of 16 rows

  // ---- precompute (t-independent) ----
  k_zero_f32<<<gN64, B, 0, stream>>>(out, N64);
  k_zero_f32<<<gNodes, B, 0, stream>>>(m1f, N);
  k_gemm_n64<<<gGemm, B, 0, stream>>>(x, W1, H1, N);
  k_node_alpha<<<gNodes, B, 0, stream>>>(H1, a1s, a1d, al1s, al1d, N);
  k_edge_logit_max<<<gEdges, B, 0, stream>>>(src1, dst1, al1s, al1d, e1base,
                                             (unsigned*)m1f, E);
  k_finalize_max<<<gNodes, B, 0, stream>>>((unsigned*)m1f, N);

  // ---- time loop: ts = linspace(0,1,10), hstep = 1/9 ----
  const int   STEPS = 10;
  const float hstep = 1.0f / (STEPS - 1);
  for (int k = 0; k < STEPS; ++k) {
    const float t = (float)k * hstep;

    k_zero_f32<<<cdiv(perT, B), B, 0, stream>>>(denom1, perT);

    // layer 1: w = exp(t*(e_base - m_base[dst])); num1 += w*H1[src]
    k_edge_aggr<<<gEdge16, B, 0, stream>>>(src1, dst1, e1base, m1f, H1,
                                           denom1, num1, t, t, E);
    // g = elu(t * num1 / denom1)
    k_elu_g<<<gN64, B, 0, stream>>>(num1, denom1, g, t, N64);

    // layer 2 (full attention each step)
    k_gemm_n64<<<gGemm, B, 0, stream>>>(g, W2, H2, N);
    k_node_alpha<<<gNodes, B, 0, stream>>>(H2, a2s, a2d, al2s, al2d, N);
    k_edge_logit_max<<<gEdges, B, 0, stream>>>(src2, dst2, al2s, al2d, e2,
                                               (unsigned*)m2f, E);
    k_finalize_max<<<gNodes, B, 0, stream>>>((unsigned*)m2f, N);
    k_edge_aggr<<<gEdge16, B, 0, stream>>>(src2, dst2, e2, m2f, H2,
                                           denom2, num2, 1.0f, 1.0f, E);
    // out += hstep * num2 / denom2
    k_out_accum<<<gN64, B, 0, stream>>>(out, num2, denom2, hstep, N64);
  }
}